// HGNNAnomalyDetector_69990787055765
// MI455X (gfx1250) — compile-verified
//
#include <hip/hip_runtime.h>
#include <hip/hip_bf16.h>

#define N_NODES 50000
#define N_EDGES 50000
#define NNZ     800000
#define IN_C    256
#define HID_C   128
#define OUT_C   64

typedef __attribute__((ext_vector_type(2))) float v2f;
typedef __attribute__((ext_vector_type(8))) float v8f;

// ---------------- utility kernels ----------------

__global__ void zero_f32v4(float4* __restrict__ p, long long n4) {
  long long i = (long long)blockIdx.x * blockDim.x + threadIdx.x;
  long long stride = (long long)gridDim.x * blockDim.x;
  float4 z = {0.0f, 0.0f, 0.0f, 0.0f};
  for (; i < n4; i += stride) p[i] = z;
}

__global__ void count_degrees(const int* __restrict__ nodeIdx,
                              const int* __restrict__ edgeIdx,
                              float* __restrict__ dcnt,
                              float* __restrict__ bcnt, int nnz) {
  int i = blockIdx.x * blockDim.x + threadIdx.x;
  if (i < nnz) {
    unsafeAtomicAdd(&dcnt[nodeIdx[i]], 1.0f);
    unsafeAtomicAdd(&bcnt[edgeIdx[i]], 1.0f);
  }
}

// dst[dstIdx[e]][c] += src[srcIdx[e]][c]  for all (e,c).
// Mapping i -> (e = i/C, c = i%C): all lanes of a wave share one e for C>=32,
// so the index loads are wave-uniform (scalarized).
template<int C>
__global__ void scatter_add(const float* __restrict__ src,
                            const int* __restrict__ srcIdx,
                            const int* __restrict__ dstIdx,
                            float* __restrict__ dst, int nnz) {
  long long total = (long long)nnz * C;
  long long stride = (long long)gridDim.x * blockDim.x;
  for (long long i = (long long)blockIdx.x * blockDim.x + threadIdx.x;
       i < total; i += stride) {
    int e = (int)(i / C);
    int c = (int)(i - (long long)e * C);
    int s = srcIdx[e];
    int d = dstIdx[e];
    unsafeAtomicAdd(&dst[(long long)d * C + c], src[(long long)s * C + c]);
  }
}

// data[r][c] *= (cnt[r] > 0 ? 1/cnt[r] : 0), vectorized float4.
template<int C>
__global__ void scale_rows_inv(float4* __restrict__ data,
                               const float* __restrict__ cnt, int rows) {
  constexpr int C4 = C / 4;
  long long total = (long long)rows * C4;
  long long stride = (long long)gridDim.x * blockDim.x;
  for (long long i = (long long)blockIdx.x * blockDim.x + threadIdx.x;
       i < total; i += stride) {
    int r = (int)(i / C4);
    float cv = cnt[r];
    float inv = cv > 0.0f ? 1.0f / cv : 0.0f;
    float4 v = data[i];
    v.x *= inv; v.y *= inv; v.z *= inv; v.w *= inv;
    data[i] = v;
  }
}

// dst[r][c] = relu(acc[r][c] * (1/cnt[r]) + bias[c]), vectorized float4.
template<int C>
__global__ void finalize_relu(const float4* __restrict__ acc,
                              const float* __restrict__ cnt,
                              const float4* __restrict__ bias,
                              float4* __restrict__ dst, int rows) {
  constexpr int C4 = C / 4;
  long long total = (long long)rows * C4;
  long long stride = (long long)gridDim.x * blockDim.x;
  for (long long i = (long long)blockIdx.x * blockDim.x + threadIdx.x;
       i < total; i += stride) {
    int r = (int)(i / C4);
    int c4 = (int)(i - (long long)r * C4);
    float cv = cnt[r];
    float inv = cv > 0.0f ? 1.0f / cv : 0.0f;
    float4 a = acc[i];
    float4 b = bias[c4];
    float4 o;
    o.x = a.x * inv + b.x; o.x = o.x > 0.0f ? o.x : 0.0f;
    o.y = a.y * inv + b.y; o.y = o.y > 0.0f ? o.y : 0.0f;
    o.z = a.z * inv + b.z; o.z = o.z > 0.0f ? o.z : 0.0f;
    o.w = a.w * inv + b.w; o.w = o.w > 0.0f ? o.w : 0.0f;
    dst[i] = o;
  }
}

// ---------------- WMMA fp32 GEMM ----------------
// D[M x N] = A[M x K] @ B[K x N] (+ bias if HAS_BIAS), N = NWAVES*16, lda = K.
// One block computes a 16-row slab; wave w computes the 16x16 tile at cols w*16.
// Uses V_WMMA_F32_16X16X4_F32 (exact fp32, wave32).
template<int K, int NWAVES, bool HAS_BIAS>
__global__ void gemm_wmma_f32(const float* __restrict__ A,
                              const float* __restrict__ B,
                              const float* __restrict__ bias,
                              float* __restrict__ D) {
  constexpr int N = NWAVES * 16;
  constexpr int K4 = K / 4;
  __shared__ float As[16][K + 4];   // (K+4)%64 == 4 -> conflict-free column reads

  const int tid = threadIdx.x;
  const int m0 = blockIdx.x * 16;

  // Cooperative, vectorized load of the 16xK A tile into LDS.
  const float4* Ag = reinterpret_cast<const float4*>(A + (long long)m0 * K);
  for (int i = tid; i < 16 * K4; i += NWAVES * 32) {
    int row = i / K4;
    int c4 = i - row * K4;
    float4 v = Ag[(long long)row * K4 + c4];
    *reinterpret_cast<float4*>(&As[row][c4 * 4]) = v;
  }
  __syncthreads();

  const int lane = tid & 31;
  const int wv = tid >> 5;
  const int half = lane >> 4;   // lanes 16-31 carry K=+2,+3 per ISA layout
  const int l16 = lane & 15;
  const int n = wv * 16 + l16;

  v8f acc = {};
  for (int k0 = 0; k0 < K; k0 += 4) {
    const int ka = k0 + half * 2;
    v2f a, b;
    a.x = As[l16][ka];            // A: lane = M, VGPR0/1 = K, K+1
    a.y = As[l16][ka + 1];
    const float* Bp = B + (long long)ka * N + n;
    b.x = Bp[0];                  // B: lane = N, VGPR0/1 = K, K+1
    b.y = Bp[N];
    acc = __builtin_amdgcn_wmma_f32_16x16x4_f32(
        false, a, false, b, (short)0, acc, false, false);
  }

  // Branch-free epilogue: one bias load per lane (column-invariant over rows).
  const float bv = HAS_BIAS ? bias[n] : 0.0f;
#pragma unroll
  for (int r = 0; r < 8; ++r) {
    int row = r + half * 8;       // D layout: VGPR r -> M = r (+8 for hi half)
    D[(long long)(m0 + row) * N + n] = acc[r] + bv;
  }
}

// ---------------- driver ----------------

extern "C" void kernel_launch(void* const* d_in, const int* in_sizes, int n_in,
                              void* d_out, int out_size, void* d_ws, size_t ws_size,
                              hipStream_t stream) {
  (void)in_sizes; (void)n_in; (void)out_size; (void)ws_size;

  const float* x       = (const float*)d_in[0];
  const int*   nodeIdx = (const int*)d_in[1];         // hyperedge_index row 0
  const int*   edgeIdx = nodeIdx + NNZ;               // hyperedge_index row 1
  const float* W1 = (const float*)d_in[2];
  const float* b1 = (const float*)d_in[3];
  const float* W2 = (const float*)d_in[4];
  const float* b2 = (const float*)d_in[5];
  const float* Wp = (const float*)d_in[6];
  const float* bp = (const float*)d_in[7];

  float* out = (float*)d_out;
  float* z  = out;                                    // [50000, 64]
  float* f1 = out + (long long)N_NODES * OUT_C;       // [50000, 128]
  float* f2 = f1 + (long long)N_NODES * HID_C;        // [50000, 64]

  float* ws   = (float*)d_ws;
  float* bufA = ws;                                   // 50000*128 floats
  float* bufB = bufA + (long long)N_NODES * HID_C;    // 50000*128 floats
  float* dcnt = bufB + (long long)N_NODES * HID_C;    // 50000
  float* bcnt = dcnt + N_NODES;                       // 50000

  const int TB = 256;

  // degrees (dcnt & bcnt are contiguous; 2*50000 floats = 25000 float4)
  zero_f32v4<<<128, TB, 0, stream>>>((float4*)dcnt, (2LL * N_NODES) / 4);
  count_degrees<<<(NNZ + TB - 1) / TB, TB, 0, stream>>>(nodeIdx, edgeIdx, dcnt, bcnt, NNZ);

  // ---- layer 1: h1 = relu(Dinv * H * Binv * H^T * (x@W1) + b1) ----
  gemm_wmma_f32<IN_C, HID_C / 16, false>
      <<<N_NODES / 16, (HID_C / 16) * 32, 0, stream>>>(x, W1, nullptr, bufA);   // xw1
  zero_f32v4<<<2048, TB, 0, stream>>>((float4*)bufB, (long long)N_EDGES * HID_C / 4);
  scatter_add<HID_C><<<4096, TB, 0, stream>>>(bufA, nodeIdx, edgeIdx, bufB, NNZ); // edge agg
  scale_rows_inv<HID_C><<<2048, TB, 0, stream>>>((float4*)bufB, bcnt, N_EDGES);
  zero_f32v4<<<2048, TB, 0, stream>>>((float4*)bufA, (long long)N_NODES * HID_C / 4);
  scatter_add<HID_C><<<4096, TB, 0, stream>>>(bufB, edgeIdx, nodeIdx, bufA, NNZ); // node agg
  finalize_relu<HID_C><<<2048, TB, 0, stream>>>(
      (const float4*)bufA, dcnt, (const float4*)b1, (float4*)f1, N_NODES);

  // ---- layer 2: h2 = relu(Dinv * H * Binv * H^T * (h1@W2) + b2) ----
  gemm_wmma_f32<HID_C, OUT_C / 16, false>
      <<<N_NODES / 16, (OUT_C / 16) * 32, 0, stream>>>(f1, W2, nullptr, bufB);  // xw2
  zero_f32v4<<<2048, TB, 0, stream>>>((float4*)bufA, (long long)N_EDGES * OUT_C / 4);
  scatter_add<OUT_C><<<4096, TB, 0, stream>>>(bufB, nodeIdx, edgeIdx, bufA, NNZ);
  scale_rows_inv<OUT_C><<<2048, TB, 0, stream>>>((float4*)bufA, bcnt, N_EDGES);
  zero_f32v4<<<2048, TB, 0, stream>>>((float4*)bufB, (long long)N_NODES * OUT_C / 4);
  scatter_add<OUT_C><<<4096, TB, 0, stream>>>(bufA, edgeIdx, nodeIdx, bufB, NNZ);
  finalize_relu<OUT_C><<<2048, TB, 0, stream>>>(
      (const float4*)bufB, dcnt, (const float4*)b2, (float4*)f2, N_NODES);

  // ---- projection: z = h2 @ Wp + bp ----
  gemm_wmma_f32<OUT_C, OUT_C / 16, true>
      <<<N_NODES / 16, (OUT_C / 16) * 32, 0, stream>>>(f2, Wp, bp, z);
}